// StatePredictor_71829033059116
// MI455X (gfx1250) — compile-verified
//
#include <hip/hip_runtime.h>
#include <hip/hip_bf16.h>

// ---------------------------------------------------------------------------
// Problem constants (from reference, L == 1 specialization)
// ---------------------------------------------------------------------------
#define BATCH    4096
#define DMODEL   1024
#define DINNER   2048
#define DTRANK   64
#define DSTATE   16
#define DBC_N    (DTRANK + 2 * DSTATE)   // 96
#define LN_EPS   1e-5f

typedef __attribute__((ext_vector_type(2))) float v2f;
typedef __attribute__((ext_vector_type(8))) float v8f;
typedef __attribute__((ext_vector_type(4))) int   v4i;

// ---------------------------------------------------------------------------
// CDNA5 async global->LDS staging (ASYNCcnt path), guarded so the compile
// stays clean if the toolchain spells the builtins differently.
// Probe result: builtin exists with signature taking `int __vector(4)*` args.
// ---------------------------------------------------------------------------
#ifdef __AMDGCN__
#if __has_builtin(__builtin_amdgcn_global_load_async_to_lds_b128) && \
    __has_builtin(__builtin_amdgcn_s_wait_asynccnt)
#define USE_ASYNC_LDS 1
#endif
#endif
#ifndef USE_ASYNC_LDS
#define USE_ASYNC_LDS 0
#endif

__device__ __forceinline__ void stage16(const float* __restrict__ g, float* l) {
#if USE_ASYNC_LDS
    __builtin_amdgcn_global_load_async_to_lds_b128(
        (v4i*)g, (v4i*)l, /*offset=*/0, /*cpol=*/0);
#else
    *reinterpret_cast<float4*>(l) = *reinterpret_cast<const float4*>(g);
#endif
}
__device__ __forceinline__ void stage_commit() {
#if USE_ASYNC_LDS
    __builtin_amdgcn_s_wait_asynccnt(0);
#endif
    __syncthreads();
}

__device__ __forceinline__ float silu_f(float x) {
    return x / (1.0f + expf(-x));
}
__device__ __forceinline__ float softplus_f(float x) {
    return (x > 20.0f) ? x : log1pf(expf(x));
}

// ---------------------------------------------------------------------------
// Kernel 1: LayerNorm over D_MODEL, one block per row
// ---------------------------------------------------------------------------
__global__ __launch_bounds__(256)
void ln_kernel(const float* __restrict__ x, const float* __restrict__ g,
               const float* __restrict__ b, float* __restrict__ xn) {
    __shared__ float s_sum[256];
    __shared__ float s_sq[256];
    const int row = blockIdx.x;
    const int tid = threadIdx.x;
    const float4 v = *reinterpret_cast<const float4*>(x + (size_t)row * DMODEL + tid * 4);
    s_sum[tid] = v.x + v.y + v.z + v.w;
    s_sq[tid]  = v.x * v.x + v.y * v.y + v.z * v.z + v.w * v.w;
    __syncthreads();
    #pragma unroll
    for (int ofs = 128; ofs > 0; ofs >>= 1) {
        if (tid < ofs) { s_sum[tid] += s_sum[tid + ofs]; s_sq[tid] += s_sq[tid + ofs]; }
        __syncthreads();
    }
    const float mu  = s_sum[0] * (1.0f / DMODEL);
    const float var = s_sq[0] * (1.0f / DMODEL) - mu * mu;
    const float rs  = rsqrtf(var + LN_EPS);
    float4 o;
    const int c = tid * 4;
    o.x = (v.x - mu) * rs * g[c + 0] + b[c + 0];
    o.y = (v.y - mu) * rs * g[c + 1] + b[c + 1];
    o.z = (v.z - mu) * rs * g[c + 2] + b[c + 2];
    o.w = (v.w - mu) * rs * g[c + 3] + b[c + 3];
    *reinterpret_cast<float4*>(xn + (size_t)row * DMODEL + c) = o;
}

// ---------------------------------------------------------------------------
// Kernel 3b: per-row bc = dot(Bm, Cm) over D_STATE
// dbc layout: [row][96] = [dt(64) | B(16) | C(16)]
// ---------------------------------------------------------------------------
__global__ __launch_bounds__(256)
void bc_kernel(const float* __restrict__ dbc, float* __restrict__ bc) {
    const int row = blockIdx.x * 256 + threadIdx.x;
    if (row >= BATCH) return;
    const float* p = dbc + (size_t)row * DBC_N;
    float acc = 0.0f;
    #pragma unroll
    for (int s = 0; s < DSTATE; ++s)
        acc += p[DTRANK + s] * p[DTRANK + DSTATE + s];
    bc[row] = acc;
}

// ---------------------------------------------------------------------------
// Fused epilogues
// ---------------------------------------------------------------------------
struct EpiGemm1 {           // xz -> u (cols 0..2047), sz (cols 2048..4095)
    const float* cw;        // conv_w (DINNER,4); only tap 3 survives at L==1
    const float* cb;        // conv_b
    float* u;
    float* sz;
    __device__ void operator()(int row, int col, float v) const {
        if (col < DINNER) {
            const float t = v * cw[col * 4 + 3] + cb[col];
            u[(size_t)row * DINNER + col] = silu_f(t);
        } else {
            sz[(size_t)row * DINNER + (col - DINNER)] = silu_f(v);
        }
    }
};
struct EpiStore {           // plain store, row-major with ld
    float* out; int ld;
    __device__ void operator()(int row, int col, float v) const {
        out[(size_t)row * ld + col] = v;
    }
};
struct EpiY {               // dt-gemm -> y = u*sz*(softplus(v+b_dt)*bc + D_skip)
    const float* b_dt;
    const float* dskip;
    const float* bc;
    const float* u;
    const float* sz;
    float* y;
    __device__ void operator()(int row, int col, float v) const {
        const float dt = softplus_f(v + b_dt[col]);
        const size_t i = (size_t)row * DINNER + col;
        y[i] = u[i] * sz[i] * (dt * bc[row] + dskip[col]);
    }
};

// ---------------------------------------------------------------------------
// WMMA fp32 GEMM core: C(MxN) = A(MxK, lda) * B(KxN, ldb), fused epilogue.
//  - Block tile BM x BN, K staged in chunks of 16, double-buffered LDS.
//  - A in LDS: m-major, row stride KC+4 = 20 floats (16B-aligned rows,
//    stride-20 bank walk => conflict-free b64 fragment reads).
//  - B in LDS: k-major (n-contiguous rows) => async B128-friendly.
//  - Staging uses GLOBAL_LOAD_ASYNC_TO_LDS_B128 (ASYNCcnt) when available.
// Requires: M % BM == 0, N % BN == 0, K % 16 == 0 (true for all call sites).
// ---------------------------------------------------------------------------
template <int WAVES_M, int WAVES_N, int WM, int WN, typename Epi>
__global__ __launch_bounds__(WAVES_M * WAVES_N * 32)
void gemm_wmma_f32(const float* __restrict__ A, int lda,
                   const float* __restrict__ B, int ldb,
                   int K, Epi epi) {
    constexpr int BM = WAVES_M * WM * 16;
    constexpr int BN = WAVES_N * WN * 16;
    constexpr int NT = WAVES_M * WAVES_N * 32;
    constexpr int KC = 16;
    constexpr int AP = KC + 4;   // padded A row stride (floats)

    __shared__ __align__(16) float As[2][BM][AP];
    __shared__ __align__(16) float Bs[2][KC][BN];

    const int tid  = threadIdx.x;
    const int lane = tid & 31;
    const int wid  = tid >> 5;
    const int wm   = wid % WAVES_M;
    const int wn   = wid / WAVES_M;

    const int row0 = blockIdx.x * BM;
    const int col0 = blockIdx.y * BN;

    const int mlane = lane & 15;          // M (A) / N (B) index within tile
    const int kb0   = (lane >> 4) << 1;   // lanes 0-15 -> K{0,1}, 16-31 -> K{2,3}

    v8f acc[WM][WN];
    #pragma unroll
    for (int i = 0; i < WM; ++i)
        #pragma unroll
        for (int j = 0; j < WN; ++j)
            #pragma unroll
            for (int r = 0; r < 8; ++r)
                acc[i][j][r] = 0.0f;

    auto stageA = [&](int buf, int k0) {
        for (int i = tid; i < BM * (KC / 4); i += NT) {
            const int r  = i >> 2;
            const int kq = (i & 3) << 2;
            stage16(A + (size_t)(row0 + r) * lda + k0 + kq, &As[buf][r][kq]);
        }
    };
    auto stageB = [&](int buf, int k0) {
        for (int i = tid; i < KC * (BN / 4); i += NT) {
            const int kr = i / (BN / 4);
            const int nc = (i % (BN / 4)) << 2;
            stage16(B + (size_t)(k0 + kr) * ldb + col0 + nc, &Bs[buf][kr][nc]);
        }
    };

    const int nch = K / KC;
    stageA(0, 0);
    stageB(0, 0);
    stage_commit();

    for (int c = 0; c < nch; ++c) {
        const int cur = c & 1;
        if (c + 1 < nch) {                 // prefetch next chunk into other buffer
            stageA(1 - cur, (c + 1) * KC);
            stageB(1 - cur, (c + 1) * KC);
        }
        #pragma unroll
        for (int ks = 0; ks < KC; ks += 4) {
            v2f a[WM];
            v2f b[WN];
            #pragma unroll
            for (int i = 0; i < WM; ++i) {
                const int m = wm * WM * 16 + i * 16 + mlane;
                a[i] = *reinterpret_cast<const v2f*>(&As[cur][m][ks + kb0]);
            }
            #pragma unroll
            for (int j = 0; j < WN; ++j) {
                const int n = wn * WN * 16 + j * 16 + mlane;
                b[j] = v2f{Bs[cur][ks + kb0][n], Bs[cur][ks + kb0 + 1][n]};
            }
            #pragma unroll
            for (int i = 0; i < WM; ++i)
                #pragma unroll
                for (int j = 0; j < WN; ++j)
                    acc[i][j] = __builtin_amdgcn_wmma_f32_16x16x4_f32(
                        false, a[i], false, b[j], (short)0, acc[i][j],
                        false, false);
        }
        stage_commit();   // drain this wave's asyncs for chunk c+1, then barrier
    }

    // epilogue: C layout = VGPR r -> M = r (+8 for hi lanes), lane -> N
    const int rbase = (lane >> 4) << 3;
    #pragma unroll
    for (int i = 0; i < WM; ++i)
        #pragma unroll
        for (int j = 0; j < WN; ++j) {
            const int mt = row0 + wm * WM * 16 + i * 16;
            const int nt = col0 + wn * WN * 16 + j * 16 + mlane;
            #pragma unroll
            for (int r = 0; r < 8; ++r)
                epi(mt + r + rbase, nt, acc[i][j][r]);
        }
}

// ---------------------------------------------------------------------------
// Host-side launch
// ---------------------------------------------------------------------------
extern "C" void kernel_launch(void* const* d_in, const int* in_sizes, int n_in,
                              void* d_out, int out_size, void* d_ws, size_t ws_size,
                              hipStream_t stream) {
    (void)in_sizes; (void)n_in; (void)out_size; (void)ws_size;

    const float* z_state = (const float*)d_in[0];   // (4096, 1024)
    const float* ln_g    = (const float*)d_in[1];   // (1024,)
    const float* ln_b    = (const float*)d_in[2];   // (1024,)
    const float* W_in    = (const float*)d_in[3];   // (1024, 4096)
    const float* conv_w  = (const float*)d_in[4];   // (2048, 4)
    const float* conv_b  = (const float*)d_in[5];   // (2048,)
    const float* W_x     = (const float*)d_in[6];   // (2048, 96)
    const float* W_dt    = (const float*)d_in[7];   // (64, 2048)
    const float* b_dt    = (const float*)d_in[8];   // (2048,)
    /* d_in[9]  = A_log : dead (h0 == 0 and L == 1 => dA multiplies zero) */
    const float* D_skip  = (const float*)d_in[10];  // (2048,)
    const float* W_out   = (const float*)d_in[11];  // (2048, 1024)
    float* out = (float*)d_out;                     // (4096, 1024)

    // workspace layout (floats)
    float* ws   = (float*)d_ws;
    float* buf0 = ws;                                   // xn (4096x1024) then y (4096x2048)
    float* u    = buf0 + (size_t)BATCH * DINNER;        // 4096x2048
    float* sz   = u    + (size_t)BATCH * DINNER;        // 4096x2048
    float* dbc  = sz   + (size_t)BATCH * DINNER;        // 4096x96
    float* bc   = dbc  + (size_t)BATCH * DBC_N;         // 4096

    // 1) LayerNorm
    ln_kernel<<<BATCH, 256, 0, stream>>>(z_state, ln_g, ln_b, buf0);

    // 2) xz = xn @ W_in  (4096 x 1024 x 4096), fused conv-tap/silu epilogue
    gemm_wmma_f32<4, 2, 2, 2, EpiGemm1>
        <<<dim3(BATCH / 128, (2 * DINNER) / 64), 256, 0, stream>>>(
            buf0, DMODEL, W_in, 2 * DINNER, DMODEL,
            EpiGemm1{conv_w, conv_b, u, sz});

    // 3) dbc = u @ W_x   (4096 x 2048 x 96), skinny-N config (BN = 32)
    gemm_wmma_f32<8, 1, 1, 2, EpiStore>
        <<<dim3(BATCH / 128, DBC_N / 32), 256, 0, stream>>>(
            u, DINNER, W_x, DBC_N, DINNER, EpiStore{dbc, DBC_N});

    // 3b) bc[row] = dot(B, C)
    bc_kernel<<<BATCH / 256, 256, 0, stream>>>(dbc, bc);

    // 4) dt = softplus(dbc[:, :64] @ W_dt + b_dt); y = u*sz*(dt*bc + D_skip)
    gemm_wmma_f32<4, 2, 2, 2, EpiY>
        <<<dim3(BATCH / 128, DINNER / 64), 256, 0, stream>>>(
            dbc, DBC_N, W_dt, DINNER, DTRANK,
            EpiY{b_dt, D_skip, bc, u, sz, buf0});

    // 5) out = y @ W_out (4096 x 2048 x 1024)
    gemm_wmma_f32<4, 2, 2, 2, EpiStore>
        <<<dim3(BATCH / 128, DMODEL / 64), 256, 0, stream>>>(
            buf0, DINNER, W_out, DMODEL, DINNER, EpiStore{out, DMODEL});
}